// GraphConvAttnPool_89970974916833
// MI455X (gfx1250) — compile-verified
//
#include <hip/hip_runtime.h>
#include <hip/hip_bf16.h>

// ---------------- problem constants ----------------
constexpr int NB   = 4;      // batch
constexpr int NL   = 2500;   // sequence length
constexpr int NE   = 100;    // embed dim
constexpr int NF   = 50;     // conv filters / feature dim
constexpr int NY   = 8921;   // labels
constexpr int NK   = 9;      // conv kernel
constexpr int NPAD = 4;      // NK/2
constexpr int LP   = 2512;   // NL padded to mult of 16
constexpr int LPS  = 2516;   // LDS row stride for score panel (bank-skewed)
constexpr int FP   = 64;     // NF padded to mult of 16 (K-loop friendly)
constexpr int YP   = 8960;   // NY padded to mult of 64

// ---------------- WMMA f32 16x16x4 ----------------
typedef float v2f __attribute__((ext_vector_type(2)));
typedef float v8f __attribute__((ext_vector_type(8)));

__device__ __forceinline__ v8f wmma_f32_16x16x4(v2f a, v2f b, v8f c) {
  // (neg_a, A, neg_b, B, c_mod, C, reuse_a, reuse_b)
  return __builtin_amdgcn_wmma_f32_16x16x4_f32(false, a, false, b, (short)0, c,
                                               false, false);
}

// fragment loaders: source pairs are contiguous + 8B aligned -> single b64 load
__device__ __forceinline__ v2f ldpair(const float* p) { return *(const v2f*)p; }

// ---------------- CDNA5 async global->LDS copy (ASYNCcnt) ----------------
__device__ __forceinline__ void async_ld_b128(unsigned lds_off, const float* g) {
  asm volatile("global_load_async_to_lds_b128 %0, %1, off" ::"v"(lds_off), "v"(g)
               : "memory");
}
__device__ __forceinline__ void wait_async0() {
  asm volatile("s_wait_asynccnt 0" ::: "memory");
}

// ---------------- kernel 0: zero fill ----------------
__global__ void zero_kernel(float* __restrict__ p, size_t n) {
  size_t i = (size_t)blockIdx.x * blockDim.x + threadIdx.x;
  if (i < n) p[i] = 0.0f;
}

// ---------------- kernel 1: embed + conv1d(same) + tanh ----------------
// Pair-interleaved layouts so WMMA B-fragments are single 8B loads:
//   hF2: [NB][LP/2][FP][2]  element (l,f) at ((b*LP/2 + l/2)*FP + f)*2 + (l&1)
//   hT2: [NB][FP/2][LP][2]  element (f,l) at ((b*FP/2 + f/2)*LP + l)*2 + (f&1)
__global__ void __launch_bounds__(256)
conv_embed_kernel(const int* __restrict__ tokens, const float* __restrict__ embW,
                  const float* __restrict__ convW, const float* __restrict__ convB,
                  float* __restrict__ hF2, float* __restrict__ hT2) {
  const int b  = blockIdx.y;
  const int l0 = blockIdx.x * 32;
  __shared__ float xs[40 * NE];  // positions l0-4 .. l0+35
  for (int j = threadIdx.x; j < 40 * NE; j += blockDim.x) {
    int p = j / NE, e = j % NE;
    int gl = l0 - NPAD + p;
    float v = 0.0f;
    if (gl >= 0 && gl < NL) {
      int tok = tokens[b * NL + gl];
      v = embW[(size_t)tok * NE + e];
    }
    xs[j] = v;
  }
  __syncthreads();
  const int p  = threadIdx.x & 31;
  const int fg = threadIdx.x >> 5;  // 0..7
  const int l  = l0 + p;
  for (int f = fg; f < NF; f += 8) {
    float acc = convB[f];
    const float* w = convW + (size_t)f * NE * NK;
    for (int e = 0; e < NE; ++e) {
      const float* xr = xs + p * NE + e;  // xs[(p+k)*NE + e]
#pragma unroll
      for (int k = 0; k < NK; ++k) acc = fmaf(xr[k * NE], w[e * NK + k], acc);
    }
    float hv = tanhf(acc);
    if (l < NL) {
      hF2[(((size_t)b * (LP / 2) + (l >> 1)) * FP + f) * 2 + (l & 1)] = hv;
      hT2[(((size_t)b * (FP / 2) + (f >> 1)) * LP + l) * 2 + (f & 1)] = hv;
    }
  }
}

// ---------------- kernel 2: t = label_features @ gcn_w^T ([YP][FP], zero-padded)
__global__ void gcn_t_kernel(const float* __restrict__ lf, const float* __restrict__ gw,
                             float* __restrict__ t) {
  int idx = blockIdx.x * blockDim.x + threadIdx.x;
  if (idx >= YP * FP) return;
  int y = idx / FP, f = idx % FP;
  float acc = 0.0f;
  if (y < NY && f < NF)
    for (int j = 0; j < NF; ++j) acc = fmaf(lf[y * NF + j], gw[f * NF + j], acc);
  t[idx] = acc;
}

// ---------------- kernel 3: ls = relu(adj @ t)  WMMA + async-to-LDS -----------
// block = 128 threads (4 waves), each wave owns 16 rows x all 64 cols.
// adj tile staged via GLOBAL_LOAD_ASYNC_TO_LDS_B128 (interior), t tile staged
// K-transposed so B-fragments are contiguous b64 LDS loads.
__global__ void __launch_bounds__(128)
adj_gemm_kernel(const float* __restrict__ adj, const float* __restrict__ t,
                float* __restrict__ ls) {
  __shared__ float sA[64 * 68];   // adj tile [row][k], stride 68 (bank skew)
  __shared__ float sBt[64 * 68];  // t tile transposed: [f][k], stride 68
  const int y0   = blockIdx.x * 64;
  const int lane = threadIdx.x & 31;
  const int wave = threadIdx.x >> 5;
  const int m0   = wave * 16;
  const int ml   = lane & 15;
  const int kh   = (lane >> 4) << 1;  // 0 or 2 (half-wave K offset)
  const unsigned sA_off = (unsigned)(uintptr_t)&sA[0];  // LDS byte offset

  v8f acc[4] = {};
  for (int k0 = 0; k0 < YP; k0 += 64) {
    __syncthreads();
    // ---- stage adjacency tile ----
    if ((y0 + 64 <= NY) && (k0 + 64 <= NY)) {
      // async DMA: 1024 x b128 chunks, 8 per thread
      for (int c = threadIdx.x; c < 1024; c += 128) {
        int r = c >> 4, q = (c & 15) << 2;  // row, col(floats)
        async_ld_b128(sA_off + (unsigned)((r * 68 + q) << 2),
                      adj + (size_t)(y0 + r) * NY + k0 + q);
      }
      wait_async0();
    } else {
      for (int j = threadIdx.x; j < 64 * 64; j += 128) {
        int r = j >> 6, c = j & 63;
        int gy = y0 + r, gk = k0 + c;
        sA[r * 68 + c] = (gy < NY && gk < NY) ? adj[(size_t)gy * NY + gk] : 0.0f;
      }
    }
    // ---- stage t tile transposed: sBt[f][k] = t[k0+k][f] ----
    for (int j = threadIdx.x; j < 64 * 64; j += 128) {
      int r = j >> 6, c = j & 63;  // r = k, c = f
      sBt[c * 68 + r] = t[(size_t)(k0 + r) * FP + c];
    }
    __syncthreads();
    for (int kk = 0; kk < 64; kk += 4) {
      v2f a = ldpair(&sA[(m0 + ml) * 68 + kk + kh]);
#pragma unroll
      for (int n = 0; n < 4; ++n) {
        v2f bf = ldpair(&sBt[(n * 16 + ml) * 68 + kk + kh]);
        acc[n] = wmma_f32_16x16x4(a, bf, acc[n]);
      }
    }
  }
#pragma unroll
  for (int n = 0; n < 4; ++n)
#pragma unroll
    for (int r = 0; r < 8; ++r) {
      int y = y0 + m0 + r + ((lane >> 4) << 3);
      ls[(size_t)y * FP + n * 16 + ml] = fmaxf(acc[n][r], 0.0f);
    }
}

// ---------------- kernel 4: aw = ls@attn_w^T, cw = ls@cls_w^T ([YP][FP]) -------
__global__ void label_proj_kernel(const float* __restrict__ ls,
                                  const float* __restrict__ aW,
                                  const float* __restrict__ cW,
                                  float* __restrict__ aw, float* __restrict__ cw) {
  int idx = blockIdx.x * blockDim.x + threadIdx.x;
  if (idx >= YP * FP) return;
  int y = idx / FP, f = idx % FP;
  float a = 0.0f, c = 0.0f;
  if (f < NF)
    for (int j = 0; j < NF; ++j) {
      float v = ls[(size_t)y * FP + j];
      a = fmaf(v, aW[f * NF + j], a);
      c = fmaf(v, cW[f * NF + j], c);
    }
  aw[idx] = a;
  cw[idx] = c;
}

// ---------------- kernel 5: fused attention pooling + classify -----------------
// One block per (b, 16-label tile). Full 16xL score panel lives in LDS (~157 KB,
// legal on CDNA5's 320 KB/WGP LDS) -> the 358 MB [B,Y,L] tensor never exists.
__global__ void __launch_bounds__(128)
attn_kernel(const float* __restrict__ aw, const float* __restrict__ cw,
            const float* __restrict__ hT2, const float* __restrict__ hF2,
            const float* __restrict__ bias, float* __restrict__ out) {
  extern __shared__ float sAl[];  // [16][LPS]
  __shared__ float red[16];
  const int b    = blockIdx.y;
  const int y0   = blockIdx.x * 16;
  const int lane = threadIdx.x & 31;
  const int wave = threadIdx.x >> 5;
  const int ml   = lane & 15;
  const int kh   = (lane >> 4) << 1;

  // preload A fragments: aw rows y0..y0+15, K = f in [0,64): 16 b64 loads
  v2f afr[16];
#pragma unroll
  for (int k4 = 0; k4 < 16; ++k4)
    afr[k4] = ldpair(&aw[(size_t)(y0 + ml) * FP + k4 * 4 + kh]);

  // scores[16][L] = aw_tile @ h^T   (wave w handles l-tiles w, w+4, ...)
  const float* hT2b = hT2 + (size_t)b * (FP / 2) * LP * 2;
  for (int lt = wave; lt * 16 < LP; lt += 4) {
    int l0 = lt * 16;
    v8f c = {};
#pragma unroll
    for (int k4 = 0; k4 < 16; ++k4) {
      int kk = k4 * 4 + kh;  // even
      v2f bf = ldpair(&hT2b[((size_t)(kk >> 1) * LP + l0 + ml) * 2]);
      c = wmma_f32_16x16x4(afr[k4], bf, c);
    }
#pragma unroll
    for (int r = 0; r < 8; ++r)
      sAl[(r + ((lane >> 4) << 3)) * LPS + l0 + ml] = c[r];
  }
  __syncthreads();

  // softmax over l < NL, each wave owns 4 rows; zero the padded tail
  for (int r = wave * 4; r < wave * 4 + 4; ++r) {
    float* row = sAl + r * LPS;
    float mx = -3.4e38f;
    for (int l = lane; l < NL; l += 32) mx = fmaxf(mx, row[l]);
    for (int m = 16; m >= 1; m >>= 1) mx = fmaxf(mx, __shfl_xor(mx, m, 32));
    float sum = 0.0f;
    for (int l = lane; l < NL; l += 32) {
      float e = __expf(row[l] - mx);
      row[l] = e;
      sum += e;
    }
    for (int m = 16; m >= 1; m >>= 1) sum += __shfl_xor(sum, m, 32);
    float inv = 1.0f / sum;
    for (int l = lane; l < NL; l += 32) row[l] *= inv;
    for (int l = NL + lane; l < LP; l += 32) row[l] = 0.0f;
  }
  if (threadIdx.x < 16) red[threadIdx.x] = 0.0f;
  __syncthreads();

  // m_tile[16][64] = alpha @ h  (wave w owns f-columns w*16..), then <m, cw>
  const int n0 = wave * 16;
  const float* hF2b = hF2 + (size_t)b * (LP / 2) * FP * 2;
  v8f c = {};
  for (int kk = 0; kk < LP; kk += 4) {
    int kkh = kk + kh;  // even
    v2f a  = ldpair(&sAl[ml * LPS + kkh]);
    v2f bf = ldpair(&hF2b[((size_t)(kkh >> 1) * FP + n0 + ml) * 2]);
    c = wmma_f32_16x16x4(a, bf, c);
  }
#pragma unroll
  for (int r = 0; r < 8; ++r) {
    int y = y0 + r + ((lane >> 4) << 3);
    float p = c[r] * cw[(size_t)y * FP + n0 + ml];
    p += __shfl_xor(p, 1, 32);
    p += __shfl_xor(p, 2, 32);
    p += __shfl_xor(p, 4, 32);
    p += __shfl_xor(p, 8, 32);
    if ((lane & 15) == 0) atomicAdd(&red[r + ((lane >> 4) << 3)], p);
  }
  __syncthreads();
  if (threadIdx.x < 16) {
    int y = y0 + threadIdx.x;
    if (y < NY) out[(size_t)b * NY + y] = red[threadIdx.x] + bias[y];
  }
}

// ---------------- host launcher ----------------
extern "C" void kernel_launch(void* const* d_in, const int* in_sizes, int n_in,
                              void* d_out, int out_size, void* d_ws, size_t ws_size,
                              hipStream_t stream) {
  (void)in_sizes; (void)n_in; (void)out_size; (void)ws_size;
  const int*   tokens = (const int*)d_in[0];
  const float* embW   = (const float*)d_in[1];
  const float* convW  = (const float*)d_in[2];
  const float* convB  = (const float*)d_in[3];
  const float* lf     = (const float*)d_in[4];
  const float* adj    = (const float*)d_in[5];
  const float* gw     = (const float*)d_in[6];
  const float* aW     = (const float*)d_in[7];
  const float* cW     = (const float*)d_in[8];
  const float* bias   = (const float*)d_in[9];
  float*       out    = (float*)d_out;

  float* ws  = (float*)d_ws;
  float* hF2 = ws;                                  // [NB][LP/2][FP][2]
  float* hT2 = hF2 + (size_t)NB * LP * FP;          // [NB][FP/2][LP][2]
  float* t   = hT2 + (size_t)NB * FP * LP;          // [YP][FP]
  float* ls  = t   + (size_t)YP * FP;               // [YP][FP]
  float* aw  = ls  + (size_t)YP * FP;               // [YP][FP]
  float* cw  = aw  + (size_t)YP * FP;               // [YP][FP]

  size_t nz = (size_t)2 * NB * LP * FP;
  zero_kernel<<<(unsigned)((nz + 255) / 256), 256, 0, stream>>>(hF2, nz);

  conv_embed_kernel<<<dim3((NL + 31) / 32, NB), 256, 0, stream>>>(
      tokens, embW, convW, convB, hF2, hT2);

  gcn_t_kernel<<<(YP * FP + 255) / 256, 256, 0, stream>>>(lf, gw, t);

  adj_gemm_kernel<<<YP / 64, 128, 0, stream>>>(adj, t, ls);

  label_proj_kernel<<<(YP * FP + 255) / 256, 256, 0, stream>>>(ls, aW, cW, aw, cw);

  size_t smem = (size_t)16 * LPS * sizeof(float);  // ~157.3 KB dynamic LDS
  attn_kernel<<<dim3((NY + 15) / 16, NB), 128, smem, stream>>>(
      aw, cw, hT2, hF2, bias, out);
}